// NeuralRDE_9646496546967
// MI455X (gfx1250) — compile-verified
//
#include <hip/hip_runtime.h>

// ---------------------------------------------------------------------------
// Neural RDE midpoint scan for MI455X (gfx1250), wave32 + V_WMMA_F32_16X16X4_F32
// N=1024, L=1024 (1025 steps), H=HH=D=64, I=6, O=1.
// 32 batch rows per persistent 256-thread workgroup (2 row-tiles share every
// W3 B-fragment -> half the L2 traffic); W0/W1/W2 fragments live in registers.
// Cross-lane reduction of the fused ls-contraction via shfl_xor butterfly.
// ---------------------------------------------------------------------------

typedef float v2f __attribute__((ext_vector_type(2)));
typedef float v8f __attribute__((ext_vector_type(8)));

#define WMMA_F32X4(a, b, c) \
  __builtin_amdgcn_wmma_f32_16x16x4_f32(false, (a), false, (b), (short)0, (c), false, false)

constexpr int L_TOT = 1024;
constexpr int D_DIM = 64;
constexpr int RT    = 2;          // 16-row tiles per workgroup
constexpr int ROWS  = 16 * RT;    // 32 batch rows per workgroup
constexpr int NBLK  = 1024 / ROWS;
constexpr int LSTR  = 68;         // padded LDS row stride (floats)

struct Smem {
  float h [ROWS * LSTR];   // recurrent state
  float a [ROWS * LSTR];   // scratch / k2 input
  float b [ROWS * LSTR];   // layer ping-pong
  float k [ROWS * LSTR];   // cell output (k1/k2)
  float ls[ROWS * LSTR];   // logsig tile for this step
};

// y = act(x @ W + b) for this wave's (row-tile, col-tile) job.
// W B-fragments are pre-cached in registers (wc). All 8 waves active.
__device__ __forceinline__ void small_layer(const float* X, const v2f (&wc)[16],
                                            const float* bias, float* Y, int act,
                                            int rt_s, int ct_s, int m16, int kh) {
  const int col  = ct_s * 16 + m16;
  const int arow = rt_s * 16 + m16;
  v2f av[16];
#pragma unroll
  for (int kk = 0; kk < 16; ++kk)
    av[kk] = *(const v2f*)&X[arow * LSTR + kk * 4 + kh * 2];   // ds_load_b64
  v8f acc = {};
#pragma unroll
  for (int kk = 0; kk < 16; ++kk)
    acc = WMMA_F32X4(av[kk], wc[kk], acc);
  const float bb = bias[col];
#pragma unroll
  for (int r = 0; r < 8; ++r) {
    float v = acc[r] + bb;
    v = act ? tanhf(v) : fmaxf(v, 0.0f);
    Y[(rt_s * 16 + r + 8 * kh) * LSTR + col] = v;
  }
  __syncthreads();
}

// Full vector field + fused logsig contraction: Kout[n,h] = sum_d vf(X)[n,h,d]*ls[n,d]
__device__ __forceinline__ void cell_eval(const float* X, Smem& sm,
                                          const v2f (&wc0)[16], const float* b0,
                                          const v2f (&wc1)[16], const float* b1,
                                          const v2f (&wc2)[16], const float* b2,
                                          const float* __restrict__ W3,
                                          const float* __restrict__ b3,
                                          const float (&lsreg)[RT][4][8],
                                          float* Kout,
                                          int wave, int m16, int kh, int rt_s, int ct_s) {
  small_layer(X,    wc0, b0, sm.b, 0, rt_s, ct_s, m16, kh);   // relu
  small_layer(sm.b, wc1, b1, sm.a, 0, rt_s, ct_s, m16, kh);   // relu
  small_layer(sm.a, wc2, b2, sm.b, 1, rt_s, ct_s, m16, kh);   // tanh -> z in sm.b

  // Preload A fragments (tanh z) for both row tiles; reused by all 32 col tiles.
  v2f areg[RT][16];
#pragma unroll
  for (int rt = 0; rt < RT; ++rt)
#pragma unroll
    for (int kk = 0; kk < 16; ++kk)
      areg[rt][kk] = *(const v2f*)&sm.b[(rt * 16 + m16) * LSTR + kk * 4 + kh * 2];

  // Big layer: z[32,64] @ W3[64,4096] fused with the ls contraction.
  // Column tile col = h*64 + tt*16: wave owns h = hh*8 + wave. Barrier-free.
  for (int hh = 0; hh < 8; ++hh) {
    const int h = hh * 8 + wave;
    float psum[RT][8];
#pragma unroll
    for (int rt = 0; rt < RT; ++rt)
#pragma unroll
      for (int r = 0; r < 8; ++r) psum[rt][r] = 0.0f;

#pragma unroll
    for (int tt = 0; tt < 4; ++tt) {
      const int col = h * 64 + tt * 16 + m16;
      // Phase 1: batch-load B fragments (clause -> memory-level parallelism)
      v2f bv[16];
#pragma unroll
      for (int kk = 0; kk < 16; ++kk) {
        const int kb = kk * 4 + kh * 2;
        bv[kk].x = W3[kb * 4096 + col];          // L2-resident (1 MB of 192 MB)
        bv[kk].y = W3[(kb + 1) * 4096 + col];
      }
      const float bb = b3[col];
      // Phase 2: two independent WMMA chains (one per row tile) share bv.
#pragma unroll
      for (int rt = 0; rt < RT; ++rt) {
        v8f acc = {};
#pragma unroll
        for (int kk = 0; kk < 16; ++kk)
          acc = WMMA_F32X4(areg[rt][kk], bv[kk], acc);
#pragma unroll
        for (int r = 0; r < 8; ++r)
          psum[rt][r] += (acc[r] + bb) * lsreg[rt][tt][r];   // * ls[n, d]
      }
    }

    // Sum the 16 lanes of each half (kh) with a shfl_xor butterfly.
#pragma unroll
    for (int rt = 0; rt < RT; ++rt)
#pragma unroll
      for (int r = 0; r < 8; ++r) {
        float p = psum[rt][r];
        p += __shfl_xor(p, 1);
        p += __shfl_xor(p, 2);
        p += __shfl_xor(p, 4);
        p += __shfl_xor(p, 8);
        psum[rt][r] = p;
      }
    // Lane m16<8 stores row (rt*16 + m16 + 8*kh); select psum[m16] via cndmask tree.
#pragma unroll
    for (int rt = 0; rt < RT; ++rt) {
      float v = psum[rt][0];
#pragma unroll
      for (int r = 1; r < 8; ++r) v = ((m16 & 7) == r) ? psum[rt][r] : v;
      if (m16 < 8) Kout[(rt * 16 + m16 + 8 * kh) * LSTR + h] = v;
    }
  }
  __syncthreads();
}

__global__ __launch_bounds__(256, 1)
void nrde_persistent(const float* __restrict__ initial,
                     const float* __restrict__ logsig,
                     const float* __restrict__ W_init, const float* __restrict__ b_init,
                     const float* __restrict__ W0, const float* __restrict__ b0,
                     const float* __restrict__ W1, const float* __restrict__ b1,
                     const float* __restrict__ W2, const float* __restrict__ b2,
                     const float* __restrict__ W3, const float* __restrict__ b3,
                     const float* __restrict__ Wf, const float* __restrict__ bf,
                     float* __restrict__ out) {
  __shared__ Smem sm;
  const int tid  = threadIdx.x;
  const int n0   = blockIdx.x * ROWS;
  const int wave = tid >> 5;
  const int lane = tid & 31;
  const int m16  = lane & 15;
  const int kh   = lane >> 4;
  const int rt_s = wave >> 2;   // small-layer row tile for this wave
  const int ct_s = wave & 3;    // small-layer col tile for this wave

  // Cache W0/W1/W2 B-fragments in registers for the whole kernel (weights are
  // constant across all 1025 steps -> zero global loads in 6 of 8 GEMM stages).
  v2f wc0[16], wc1[16], wc2[16];
  {
    const int col = ct_s * 16 + m16;
#pragma unroll
    for (int kk = 0; kk < 16; ++kk) {
      const int kb = kk * 4 + kh * 2;
      wc0[kk].x = W0[kb * 64 + col]; wc0[kk].y = W0[(kb + 1) * 64 + col];
      wc1[kk].x = W1[kb * 64 + col]; wc1[kk].y = W1[(kb + 1) * 64 + col];
      wc2[kk].x = W2[kb * 64 + col]; wc2[kk].y = W2[(kb + 1) * 64 + col];
    }
  }

  // h0 = initial @ W_init + b_init   (fan_in = 6)
  for (int idx = tid; idx < ROWS * 64; idx += 256) {
    const int n = idx >> 6, c = idx & 63;
    float acc = b_init[c];
#pragma unroll
    for (int i = 0; i < 6; ++i)
      acc += initial[(n0 + n) * 6 + i] * W_init[i * 64 + c];
    sm.h[n * LSTR + c] = acc;
  }
  __syncthreads();

  for (int j = 0; j <= L_TOT; ++j) {
    const int jj = (j < L_TOT - 1) ? j : (L_TOT - 1);   // last piece repeated once

    // Stage this step's logsig tile [32,64] (coalesced rows; HBM stream ~11us total)
    for (int idx = tid; idx < ROWS * 64; idx += 256) {
      const int n = idx >> 6, d = idx & 63;
      sm.ls[n * LSTR + d] = logsig[((size_t)(n0 + n) * L_TOT + jj) * D_DIM + d];
    }
    {
      const int j2 = (j + 1 < L_TOT - 1) ? (j + 1) : (L_TOT - 1);
      const int n = tid >> 3, d8 = (tid & 7) * 8;
      if (n < ROWS)
        __builtin_prefetch(&logsig[((size_t)(n0 + n) * L_TOT + j2) * D_DIM + d8], 0, 0);
    }
    __syncthreads();

    // Register-cache ls factors used in the fused contraction:
    // lsreg[rt][tt][r] = ls[row = rt*16 + r + 8*kh, d = tt*16 + m16]
    float lsreg[RT][4][8];
#pragma unroll
    for (int rt = 0; rt < RT; ++rt)
#pragma unroll
      for (int tt = 0; tt < 4; ++tt)
#pragma unroll
        for (int r = 0; r < 8; ++r)
          lsreg[rt][tt][r] = sm.ls[(rt * 16 + r + 8 * kh) * LSTR + tt * 16 + m16];

    // k1 = cell(h, ls)
    cell_eval(sm.h, sm, wc0, b0, wc1, b1, wc2, b2, W3, b3, lsreg, sm.k,
              wave, m16, kh, rt_s, ct_s);

    // midpoint input: a = h + 0.5*k1
    for (int idx = tid; idx < ROWS * 64; idx += 256) {
      const int n = idx >> 6, c = idx & 63;
      sm.a[n * LSTR + c] = sm.h[n * LSTR + c] + 0.5f * sm.k[n * LSTR + c];
    }
    __syncthreads();

    // k2 = cell(h + 0.5*k1, ls)
    cell_eval(sm.a, sm, wc0, b0, wc1, b1, wc2, b2, W3, b3, lsreg, sm.k,
              wave, m16, kh, rt_s, ct_s);

    // h += k2
    for (int idx = tid; idx < ROWS * 64; idx += 256) {
      const int n = idx >> 6, c = idx & 63;
      sm.h[n * LSTR + c] += sm.k[n * LSTR + c];
    }
    __syncthreads();
  }

  // out[n] = h[n,:] @ Wf + bf
  if (tid < ROWS) {
    float acc = bf[0];
#pragma unroll
    for (int c = 0; c < 64; ++c) acc += sm.h[tid * LSTR + c] * Wf[c];
    out[n0 + tid] = acc;
  }
}

extern "C" void kernel_launch(void* const* d_in, const int* in_sizes, int n_in,
                              void* d_out, int out_size, void* d_ws, size_t ws_size,
                              hipStream_t stream) {
  (void)in_sizes; (void)n_in; (void)out_size; (void)d_ws; (void)ws_size;
  const float* initial = (const float*)d_in[0];
  const float* logsig  = (const float*)d_in[1];
  const float* W_init  = (const float*)d_in[2];
  const float* b_init  = (const float*)d_in[3];
  const float* W0 = (const float*)d_in[4];
  const float* b0 = (const float*)d_in[5];
  const float* W1 = (const float*)d_in[6];
  const float* b1 = (const float*)d_in[7];
  const float* W2 = (const float*)d_in[8];
  const float* b2 = (const float*)d_in[9];
  const float* W3 = (const float*)d_in[10];
  const float* b3 = (const float*)d_in[11];
  const float* Wf = (const float*)d_in[12];
  const float* bf = (const float*)d_in[13];
  float* out = (float*)d_out;

  hipLaunchKernelGGL(nrde_persistent, dim3(NBLK), dim3(256), 0, stream,
                     initial, logsig, W_init, b_init,
                     W0, b0, W1, b1, W2, b2, W3, b3, Wf, bf, out);
}